// HyperbolicParamAdapter_5952824673026
// MI455X (gfx1250) — compile-verified
//
#include <hip/hip_runtime.h>
#include <stdint.h>

typedef float v2f __attribute__((ext_vector_type(2)));
typedef float v4f __attribute__((ext_vector_type(4)));
typedef float v8f __attribute__((ext_vector_type(8)));

#define SC    0.1f     /* sqrt(curvature), c = 0.01 */
#define MINN  1e-5f
#define EPSC  1e-5f

// One wave (32 threads) per block; each wave owns 16 rows of 512 f32.
// out[i,:] = alpha_i * (w .* u[i,:]) with alpha_i a function of
//   S1 = sum_j u_ij^2  and  S2 = sum_j (u_ij*w_j)^2,
// both computed simultaneously by chained V_WMMA_F32_16X16X4_F32.
// HBM -> LDS staging uses GLOBAL_LOAD_ASYNC_TO_LDS_B128 (ASYNCcnt path).
__launch_bounds__(32)
__global__ void hyp_adapter_kernel(const float* __restrict__ params,
                                   const float* __restrict__ weights,
                                   float* __restrict__ out,
                                   int nrows) {
    __shared__ float tile[16 * 516];   // 16 rows, stride 516 floats (2064 B)
    __shared__ float wsh[512];
    __shared__ float s1a[16];
    __shared__ float s2a[16];
    __shared__ float alpha[16];

    const int lane = threadIdx.x;          // 0..31 (wave32)
    const int row0 = blockIdx.x * 16;

    // ---- stage weights into LDS ----
    #pragma unroll
    for (int i = 0; i < 16; ++i)
        wsh[lane + i * 32] = weights[lane + i * 32];

    // ---- Phase 1: stream 16x512 f32 tile HBM -> LDS via async-to-LDS ----
    // flat float4 index f4: global byte off = f4*16 ; LDS byte off = f4*16 + row*16
    {
        const uint32_t ldsbase = (uint32_t)(uintptr_t)(&tile[0]); // low 32 bits = LDS offset
        const uint64_t gbase   = (uint64_t)(uintptr_t)(params + (size_t)row0 * 512);
        for (int t = 0; t < 64; ++t) {
            int f4  = t * 32 + lane;
            int row = f4 >> 7;                       // 128 float4 per row
            uint32_t goff = (uint32_t)f4 << 4;
            uint32_t ldsa = ldsbase + goff + ((uint32_t)row << 4);
            if (row0 + row < nrows) {
                asm volatile("global_load_async_to_lds_b128 %0, %1, %2"
                             :: "v"(ldsa), "v"(goff), "s"(gbase) : "memory");
            } else {
                v4f z = {0.f, 0.f, 0.f, 0.f};
                *(v4f*)(&tile[row * 516 + ((f4 & 127) << 2)]) = z;
            }
        }
        asm volatile("s_wait_asynccnt 0x0" ::: "memory");
    }
    __syncthreads();

    // ---- Phase 2: dual reduction via WMMA f32 16x16x4 ----
    // A layout (ISA 7.12.2, 32-bit A 16x4): lane l holds row M = l&15,
    //   VGPR0 = K = 2*(l>>4), VGPR1 = K+1.
    // B[k,n] = 1 for n<8, w[k]^2 for n>=8  ->  D[m,n<8] = S1[m], D[m,n>=8] = S2[m].
    const int  r    = lane & 15;
    const int  kb   = (lane >> 4) << 1;    // 0 or 2
    const bool lowN = (lane & 15) < 8;     // N = lane&15 for B/C/D layouts
    const float* trow = &tile[r * 516];
    const v2f one2 = {1.f, 1.f};
    v8f acc = {0.f, 0.f, 0.f, 0.f, 0.f, 0.f, 0.f, 0.f};

    #pragma unroll 4
    for (int c0 = 0; c0 < 512; c0 += 4) {
        v2f a   = *(const v2f*)(trow + c0 + kb);       // u chunk (ds_load b64)
        v2f w2  = *(const v2f*)(&wsh[c0 + kb]);        // w chunk
        v2f asq = a * a;
        v2f b   = lowN ? one2 : (w2 * w2);
        acc = __builtin_amdgcn_wmma_f32_16x16x4_f32(
                  false, asq, false, b, (short)0, acc, false, false);
    }

    // ---- extract S1/S2 from C/D layout: VGPR p -> M=p (lanes 0-15), M=p+8 ----
    const int hi = lane >> 4;
    if ((lane & 15) == 0) {                // N==0 column carries S1
        #pragma unroll
        for (int p = 0; p < 8; ++p) s1a[p + 8 * hi] = acc[p];
    } else if ((lane & 15) == 8) {         // N==8 column carries S2
        #pragma unroll
        for (int p = 0; p < 8; ++p) s2a[p + 8 * hi] = acc[p];
    }
    __syncthreads();

    // ---- per-row scalar chain (lanes 0..15, one row each) ----
    if (lane < 16) {
        float s1 = s1a[lane];
        float s2 = s2a[lane];

        float u_n  = sqrtf(s1);
        float un   = fmaxf(u_n, MINN);                 // clamped ||u||
        float sarg = SC * un;
        float scale_x = tanhf(sarg) / sarg;            // x = scale_x * u
        float x_n  = fmaxf(scale_x * u_n, MINN);       // clamped ||x||
        float sq2  = sqrtf(s2);
        float mx_n = scale_x * sq2;                    // ||x .* w|| (unclamped)

        float z1 = fminf(fmaxf(SC * x_n, -1.f + EPSC), 1.f - EPSC);
        float at = atanhf(z1);
        float g  = tanhf(mx_n / x_n * at);
        float scale_res = g / (mx_n * SC) * scale_x;   // res = scale_res*(w.*u)
        float res_n = fabsf(scale_res) * sq2;          // ||res||

        // project
        float y_n = fmaxf(res_n, MINN);
        const float maxn = (1.0f - 0.001f) / SC;
        float scale_y, yfin;
        if (y_n > maxn) { scale_y = scale_res / y_n * maxn; yfin = maxn; }
        else            { scale_y = scale_res;              yfin = res_n; }

        // logmap0
        float yn2 = fmaxf(yfin, MINN);
        float z2  = fminf(fmaxf(SC * yn2, -1.f + EPSC), 1.f - EPSC);
        float scale_out = scale_y / yn2 / SC * atanhf(z2);

        alpha[lane] = scale_out;
    }
    __syncthreads();

    // ---- Phase 3: out = alpha[row] * (w .* u), NT streaming stores ----
    for (int t = 0; t < 64; ++t) {
        int f4  = t * 32 + lane;
        int row = f4 >> 7;
        int c   = (f4 & 127) << 2;
        if (row0 + row < nrows) {
            v4f u  = *(const v4f*)(&tile[row * 516 + c]);
            v4f wv = *(const v4f*)(&wsh[c]);
            float a = alpha[row];
            v4f o = u * wv * a;
            __builtin_nontemporal_store(
                o, (v4f*)(out + (size_t)(row0 + row) * 512 + c));
        }
    }
}

extern "C" void kernel_launch(void* const* d_in, const int* in_sizes, int n_in,
                              void* d_out, int out_size, void* d_ws, size_t ws_size,
                              hipStream_t stream) {
    const float* params  = (const float*)d_in[0];
    const float* weights = (const float*)d_in[1];
    float*       out     = (float*)d_out;

    const int dim   = 512;
    const int nrows = in_sizes[0] / dim;           // 131072
    const int blocks = (nrows + 15) / 16;          // 16 rows per (single-wave) block

    hipLaunchKernelGGL(hyp_adapter_kernel, dim3(blocks), dim3(32), 0, stream,
                       params, weights, out, nrows);
}